// DRAELoss_42451456753936
// MI455X (gfx1250) — compile-verified
//
#include <hip/hip_runtime.h>

typedef __attribute__((ext_vector_type(2))) float v2f;
typedef __attribute__((ext_vector_type(4))) float f32x4;
typedef __attribute__((ext_vector_type(8))) float v8f;

#define N_ROWS 8192
#define D_COLS 4096
#define LAMBDA_W 0.1f

// ---------------------------------------------------------------------------
// Kernel 1: Err[row] = sum_j (input[row][j] - target[row][j])^2
// 8 waves / block, one row per wave. Streaming b128 NT loads (268MB total,
// single-use, > 192MB L2 so NT is correct). Uniform scalar loop counter so
// loop control is SALU-only and loads pipeline freely. Cross-lane reduction
// via V_WMMA_F32_16X16X4_F32 with B = ones (exact fp32).
// ---------------------------------------------------------------------------
__global__ __launch_bounds__(256) void drae_row_err(
    const float* __restrict__ input, const float* __restrict__ target,
    float* __restrict__ err)
{
    const int wave = threadIdx.x >> 5;       // 0..7
    const int lane = threadIdx.x & 31;
    const int row  = (blockIdx.x << 3) + wave;

    const f32x4* __restrict__ a4 = (const f32x4*)(input  + (size_t)row * D_COLS);
    const f32x4* __restrict__ b4 = (const f32x4*)(target + (size_t)row * D_COLS);

    float acc = 0.0f;
    const int iters = (D_COLS / 4) / 32;     // 1024 float4 / 32 lanes = 32
#pragma unroll 8
    for (int it = 0; it < iters; ++it) {     // uniform trip count: SALU loop
        const int i = it * 32 + lane;        // lane-contiguous -> coalesced
        f32x4 a = __builtin_nontemporal_load(&a4[i]);
        f32x4 b = __builtin_nontemporal_load(&b4[i]);
        f32x4 d = a - b;
        acc = fmaf(d.x, d.x, acc);
        acc = fmaf(d.y, d.y, acc);
        acc = fmaf(d.z, d.z, acc);
        acc = fmaf(d.w, d.w, acc);
    }

    // A (16x4 f32): VGPR0 = lane partial (-> A[m][0] from lane m, A[m][2] from
    // lane m+16 per documented layout), VGPR1 = 0. B = all-ones (layout
    // independent). D[m][n] = partial[m] + partial[m+16], uniform over n.
    v2f A; A.x = acc;  A.y = 0.0f;
    v2f B; B.x = 1.0f; B.y = 1.0f;
    v8f C = {};
    v8f Dm = __builtin_amdgcn_wmma_f32_16x16x4_f32(
        /*neg_a=*/false, A, /*neg_b=*/false, B,
        /*c_mod=*/(short)0, C, /*reuse_a=*/false, /*reuse_b=*/false);

    // lanes 0-15 hold D[M=0..7][*], lanes 16-31 hold D[M=8..15][*]
    float s = 0.0f;
#pragma unroll
    for (int i = 0; i < 8; ++i) s += Dm[i];
    float total = s + __shfl_xor(s, 16, 32);
    if (lane == 0) err[row] = total;
}

// ---------------------------------------------------------------------------
// Kernel 2: sort Err, prefix sums, objective scan, argmin, final scalar.
// One workgroup of 1024 threads, everything in LDS (44 KB of 320 KB/WGP).
// ---------------------------------------------------------------------------
__global__ __launch_bounds__(1024) void drae_select(
    const float* __restrict__ err, float* __restrict__ out)
{
    __shared__ float s_data[N_ROWS];   // 32 KB
    __shared__ float s_sum[1024];      // chunk sums -> scan -> min reduce
    __shared__ float s_sq[1024];       // chunk sq sums -> scan -> c1 carry
    __shared__ int   s_arg[1024];

    const int tid = threadIdx.x;

    for (int i = tid; i < N_ROWS; i += 1024) s_data[i] = err[i];
    __syncthreads();

    // Bitonic sort, ascending.
    for (int k = 2; k <= N_ROWS; k <<= 1) {
        for (int j = k >> 1; j > 0; j >>= 1) {
            for (int t = tid; t < N_ROWS; t += 1024) {
                int ixj = t ^ j;
                if (ixj > t) {
                    float va = s_data[t], vb = s_data[ixj];
                    bool up = ((t & k) == 0);
                    if (up ? (va > vb) : (va < vb)) {
                        s_data[t] = vb; s_data[ixj] = va;
                    }
                }
            }
            __syncthreads();
        }
    }

    // 8 consecutive sorted elements per thread: local sums.
    const int base = tid * 8;
    float lsum = 0.0f, lsq = 0.0f;
#pragma unroll
    for (int e = 0; e < 8; ++e) {
        float v = s_data[base + e];
        lsum += v;
        lsq = fmaf(v, v, lsq);
    }
    s_sum[tid] = lsum; s_sq[tid] = lsq;
    __syncthreads();

    // Inclusive Hillis-Steele scan over the 1024 chunk totals.
    for (int off = 1; off < 1024; off <<= 1) {
        float vs = 0.0f, vq = 0.0f;
        if (tid >= off) { vs = s_sum[tid - off]; vq = s_sq[tid - off]; }
        __syncthreads();
        s_sum[tid] += vs; s_sq[tid] += vq;
        __syncthreads();
    }

    const float total    = s_sum[1023];
    const float total_sq = s_sq[1023];
    const float excl_sum = s_sum[tid] - lsum;
    const float excl_sq  = s_sq[tid]  - lsq;

    const float nf = (float)N_ROWS;
    const float all_mean = total / nf;
    const float Sb = total_sq - nf * all_mean * all_mean;

    float best_obj = __builtin_inff();
    int   best_idx = 0x7FFFFFFF;
    float best_c1  = 0.0f;
    float c1 = excl_sum, q1 = excl_sq;
#pragma unroll
    for (int e = 0; e < 8; ++e) {
        int idx = base + e;
        float v = s_data[idx];
        c1 += v;
        q1 = fmaf(v, v, q1);
        if (idx < N_ROWS - 1) {             // obj defined for n1 = 1..N-1
            float n1 = (float)(idx + 1);
            float n2 = nf - n1;
            float mean1 = c1 / n1;
            float mean2 = (total - c1) / n2;
            float Sw1 = q1 - n1 * mean1 * mean1;
            float Sw2 = (total_sq - q1) - n2 * mean2 * mean2;
            float obj = (Sw1 + Sw2) / Sb;
            if (obj < best_obj) { best_obj = obj; best_idx = idx; best_c1 = c1; }
        }
    }
    __syncthreads();    // done reading totals; safe to reuse s_sum/s_sq

    s_sum[tid] = best_obj; s_arg[tid] = best_idx; s_sq[tid] = best_c1;
    __syncthreads();
    for (int off = 512; off > 0; off >>= 1) {
        if (tid < off) {
            float o2 = s_sum[tid + off];
            int   i2 = s_arg[tid + off];
            if (o2 < s_sum[tid] || (o2 == s_sum[tid] && i2 < s_arg[tid])) {
                s_sum[tid] = o2; s_arg[tid] = i2; s_sq[tid] = s_sq[tid + off];
            }
        }
        __syncthreads();
    }

    if (tid == 0) {
        float opt = s_sum[0];
        float T = (float)(s_arg[0] + 1);
        out[0] = s_sq[0] / T + LAMBDA_W * opt;   // SIZE_AVERAGE = True
    }
}

// ---------------------------------------------------------------------------
extern "C" void kernel_launch(void* const* d_in, const int* in_sizes, int n_in,
                              void* d_out, int out_size, void* d_ws, size_t ws_size,
                              hipStream_t stream) {
    (void)in_sizes; (void)n_in; (void)out_size; (void)ws_size;
    const float* input  = (const float*)d_in[0];
    const float* target = (const float*)d_in[1];
    float* err = (float*)d_ws;                  // 8192 floats of scratch

    drae_row_err<<<N_ROWS / 8, 256, 0, stream>>>(input, target, err);
    drae_select<<<1, 1024, 0, stream>>>(err, (float*)d_out);
}